// Dynamic_conv_39788577030872
// MI455X (gfx1250) — compile-verified
//
#include <hip/hip_runtime.h>

typedef __attribute__((ext_vector_type(2))) float v2f;
typedef __attribute__((ext_vector_type(8))) float v8f;

#define IMG_H 512
#define IMG_W 512
#define KS    15
#define PADR  7
#define NSAMP 16
#define NCH   16

__device__ __forceinline__ int reflect_idx(int i) {
    // jnp.pad mode='reflect' (no edge repeat), pad=7 < 512 so one reflection suffices
    i = (i < 0) ? -i : i;
    i = (i >= IMG_H) ? (2 * IMG_H - 2 - i) : i;
    return i;
}

// One workgroup = 8 waves = 256 threads; one sample x one 16x16 output tile.
// Each wave computes 2 output rows x 16 cols x 16 channels via 60 f32 WMMAs each.
__launch_bounds__(256)
__global__ void conv15_wmma_f32(const float* __restrict__ x,
                                const float* __restrict__ ker,
                                float* __restrict__ out,
                                float* __restrict__ smax)
{
    __shared__ float xs[30 * 32];    // 30 rows x 31 cols (stride 32) image patch
    __shared__ float wl[120 * 32];   // 120 K-pairs x (16 ch x 2) padded weights
    __shared__ float redbuf[8];

    const int n  = blockIdx.z;           // sample
    const int th = blockIdx.y * 16;      // tile row base
    const int tw = blockIdx.x * 16;      // tile col base
    const int tid = threadIdx.x;

    // only channel 0 of sample n is used by the reference
    const float* img = x + (size_t)n * (NCH * IMG_H * IMG_W);
    const float* kn  = ker + (size_t)n * (NCH * KS * KS);

    // ---- stage image patch with reflect mapping (30 x 31 valid) ----
    for (int idx = tid; idx < 30 * 31; idx += 256) {
        int lr = idx / 31;
        int lc = idx % 31;
        int gr = reflect_idx(th + lr - PADR);
        int gc = reflect_idx(tw + lc - PADR);
        xs[lr * 32 + lc] = img[gr * IMG_W + gc];
    }
    // ---- stage weights, kernel rows padded 15 -> 16 taps (zero at kx==15) ----
    // tap t = ky*16 + kx ; pair layout: wl[(t>>1)*32 + o*2 + (t&1)]
    for (int idx = tid; idx < 240 * 16; idx += 256) {
        int t  = idx >> 4;      // 0..239
        int o  = idx & 15;
        int ky = t >> 4;
        int kx = t & 15;
        float wv = (kx < 15) ? kn[o * (KS * KS) + ky * KS + kx] : 0.0f;
        wl[(t >> 1) * 32 + o * 2 + (t & 1)] = wv;
    }
    __syncthreads();

    const int lane = tid & 31;
    const int wid  = tid >> 5;
    const int j    = lane & 15;    // N index (pixel col) / M index (channel) per operand
    const int half = lane >> 4;    // selects K offset 0 or 2 within a K=4 chunk

    const int r0 = wid;            // output rows handled by this wave
    const int r1 = wid + 8;

    v8f acc0 = {};
    v8f acc1 = {};

    for (int ky = 0; ky < 15; ++ky) {
        const float* xr0 = &xs[(r0 + ky) * 32 + j];
        const float* xr1 = &xs[(r1 + ky) * 32 + j];
        const float* wrow = &wl[ky * 8 * 32 + (j << 1)];  // 8 pairs per ky
#pragma unroll
        for (int c = 0; c < 4; ++c) {
            const int kb = c * 4 + half * 2;    // this lane's K base within the chunk
            // A: 16x4 f32 weights. pair index (kb>>1) is even-aligned -> one b64 LDS load
            v2f a = *(const v2f*)&wrow[((kb >> 1)) * 32];
            // B: 4x16 f32 image patch, same half->K mapping as A
            v2f b0; b0.x = xr0[kb]; b0.y = xr0[kb + 1];
            v2f b1; b1.x = xr1[kb]; b1.y = xr1[kb + 1];
            acc0 = __builtin_amdgcn_wmma_f32_16x16x4_f32(
                false, a, false, b0, (short)0, acc0, false, false);
            acc1 = __builtin_amdgcn_wmma_f32_16x16x4_f32(
                false, a, false, b1, (short)0, acc1, false, false);
        }
    }

    // ---- store (transposed layout out[o][n][h][w]) + per-sample max ----
    float m = -1e30f;
#pragma unroll
    for (int e = 0; e < 8; ++e) {
        const int o = e + half * 8;                 // C/D layout: VGPR e, lane half
        const size_t base = ((size_t)o * NSAMP + n) * IMG_H;
        out[(base + th + r0) * IMG_W + tw + j] = acc0[e];
        out[(base + th + r1) * IMG_W + tw + j] = acc1[e];
        m = fmaxf(m, fmaxf(acc0[e], acc1[e]));
    }
#pragma unroll
    for (int off = 16; off > 0; off >>= 1)
        m = fmaxf(m, __shfl_xor(m, off, 32));
    if (lane == 0) redbuf[wid] = m;
    __syncthreads();
    if (tid == 0) {
        float bm = redbuf[0];
#pragma unroll
        for (int i2 = 1; i2 < 8; ++i2) bm = fmaxf(bm, redbuf[i2]);
        atomicMax(&smax[n], bm);
    }
}

__global__ void init_max_k(float* smax)
{
    if (threadIdx.x < NSAMP) smax[threadIdx.x] = -1e30f;
}

__launch_bounds__(256)
__global__ void normalize_k(float* __restrict__ out,
                            const float* __restrict__ smax)
{
    const int i = blockIdx.x * 256 + threadIdx.x;
    const int n = (i >> 18) & 15;      // 512*512 = 2^18 elements per (o,n) image
    out[i] = out[i] / smax[n];
}

extern "C" void kernel_launch(void* const* d_in, const int* in_sizes, int n_in,
                              void* d_out, int out_size, void* d_ws, size_t ws_size,
                              hipStream_t stream)
{
    const float* x   = (const float*)d_in[0];   // (16,16,512,512) f32
    const float* ker = (const float*)d_in[1];   // (16,16,15,15)  f32
    float* out  = (float*)d_out;                // (16,16,512,512) f32, [o][n][h][w]
    float* smax = (float*)d_ws;                 // 16 per-sample maxima

    init_max_k<<<1, 32, 0, stream>>>(smax);

    dim3 grid(IMG_W / 16, IMG_H / 16, NSAMP);   // 32 x 32 x 16 workgroups
    conv15_wmma_f32<<<grid, 256, 0, stream>>>(x, ker, out, smax);

    const int total = NCH * NSAMP * IMG_H * IMG_W;   // 67,108,864
    normalize_k<<<total / 256, 256, 0, stream>>>(out, smax);
}